// TokenSampler_65867618452182
// MI455X (gfx1250) — compile-verified
//
#include <hip/hip_runtime.h>

// TokenSampler for MI455X (gfx1250): single-query attention scores via
// V_WMMA_F32_16X16X4_F32, then rank-counting instead of a full argsort.

typedef float v2f __attribute__((ext_vector_type(2)));
typedef float v8f __attribute__((ext_vector_type(8)));

#define BH    192
#define SEQ   4096
#define DIM   64
#define RSAMP 384

// ---------------------------------------------------------------------------
// Kernel 1: zero the boolean output (harness poisons d_out; we must own it).
// ---------------------------------------------------------------------------
__global__ __launch_bounds__(256) void ts_zero(float* __restrict__ out, int n) {
  int i = blockIdx.x * 256 + threadIdx.x;
  if (i < n) out[i] = 0.0f;
}

// ---------------------------------------------------------------------------
// Kernel 2: scores a[b, s] = dot(q[b,0,:], k[b,s,:]) via f32 WMMA.
//   A operand (16x4, MxK): 16 tokens x 4 dims of k  (lane = M, half-wave = K-half)
//   B operand (4x16, KxN): q chunk broadcast across all N columns
//   D (16x16 f32): every column holds the 16 token scores; extract column 0
//   (lane 0 -> M=0..7 in VGPRs 0..7, lane 16 -> M=8..15).
// grid = (BH, 2), block = 256 (8 waves). Each wave owns 256 tokens.
// k is read exactly once, 8B/lane coalesced -> HBM-bound (~201 MB total).
// ---------------------------------------------------------------------------
__global__ __launch_bounds__(256) void ts_scores(const float* __restrict__ q,
                                                 const float* __restrict__ k,
                                                 float* __restrict__ a) {
  const int b    = blockIdx.x;
  const int lane = threadIdx.x & 31;
  const int wave = blockIdx.y * 8 + (threadIdx.x >> 5);  // 0..15
  const int half = lane >> 4;                            // K-half select
  const int lm   = lane & 15;                            // M (token-in-tile)

  const float* qb = q + (size_t)b * SEQ * DIM;           // q[b, 0, :]
  const float* kb = k + (size_t)b * SEQ * DIM;
  float*       ab = a + (size_t)b * SEQ;

  // Hoist the q-broadcast B operand: 16 K-chunks of 4 dims each.
  // Layout mirrors A: VGPR0 holds K = half*2, VGPR1 holds K = half*2 + 1.
  v2f Bq[16];
#pragma unroll
  for (int c4 = 0; c4 < 16; ++c4)
    Bq[c4] = *(const v2f*)(qb + c4 * 4 + half * 2);

  for (int g = 0; g < 16; ++g) {
    const int tokBase = wave * 256 + g * 16;
    const float* krow = kb + (size_t)(tokBase + lm) * DIM;

    v8f acc = {0.f, 0.f, 0.f, 0.f, 0.f, 0.f, 0.f, 0.f};
#pragma unroll
    for (int c4 = 0; c4 < 16; ++c4) {
      v2f A = *(const v2f*)(krow + c4 * 4 + half * 2);   // k[tok, 4-dim chunk]
      acc = __builtin_amdgcn_wmma_f32_16x16x4_f32(
          /*neg_a=*/false, A, /*neg_b=*/false, Bq[c4],
          /*c_mod=*/(short)0, acc, /*reuse_a=*/false, /*reuse_b=*/false);
    }

    // Column N==0 lives in lanes 0 (M=0..7) and 16 (M=8..15).
    if (lm == 0) {
#pragma unroll
      for (int m = 0; m < 8; ++m)
        ab[tokBase + half * 8 + m] = acc[m];
    }
  }
}

// ---------------------------------------------------------------------------
// Kernel 3: rank counting. out[b, rank(i)+1] = 1 for tokens i = 1..384;
// out[b, 0] = 1. Stable tie-break (s' < i counts as smaller) matches
// jnp.argsort. Score row staged in LDS (16 KB of the 320 KB WGP pool);
// all threads stream the same LDS address per iteration -> broadcast reads.
// ---------------------------------------------------------------------------
__global__ __launch_bounds__(256) void ts_rank(const float* __restrict__ a,
                                               float* __restrict__ out) {
  __shared__ float sa[SEQ];
  const int b = blockIdx.x;
  const float* ab = a + (size_t)b * SEQ;

  for (int s = threadIdx.x; s < SEQ; s += 256)
    sa[s] = (s == 0) ? __builtin_inff() : ab[s];  // token 0 never counted
  __syncthreads();

  float* ob = out + (size_t)b * SEQ;
  if (threadIdx.x == 0) ob[0] = 1.0f;             // prepended zero index

  for (int i = (int)threadIdx.x; i < RSAMP; i += 256) {
    const int   ti = i + 1;                       // token index 1..384
    const float v  = sa[ti];
    int rank = 0;
#pragma unroll 4
    for (int s = 0; s < SEQ; ++s) {
      const float x = sa[s];
      rank += (int)((x < v) | ((x == v) & (s < ti)));
    }
    ob[rank + 1] = 1.0f;                          // ranks unique -> no races
  }
}

// ---------------------------------------------------------------------------
extern "C" void kernel_launch(void* const* d_in, const int* in_sizes, int n_in,
                              void* d_out, int out_size, void* d_ws, size_t ws_size,
                              hipStream_t stream) {
  (void)in_sizes; (void)n_in; (void)out_size; (void)ws_size;

  const float* q = (const float*)d_in[0];
  const float* k = (const float*)d_in[1];
  float* out = (float*)d_out;
  float* a   = (float*)d_ws;                      // BH*SEQ floats = 3 MiB

  const int n = BH * SEQ;
  ts_zero<<<(n + 255) / 256, 256, 0, stream>>>(out, n);

  dim3 sgrid(BH, 2);
  ts_scores<<<sgrid, 256, 0, stream>>>(q, k, a);

  ts_rank<<<BH, 256, 0, stream>>>(a, out);
}